// GridConv_67997922230590
// MI455X (gfx1250) — compile-verified
//
#include <hip/hip_runtime.h>
#include <hip/hip_bf16.h>

typedef __attribute__((ext_vector_type(2))) float v2f;
typedef __attribute__((ext_vector_type(8))) float v8f;

#define C_IO    64
#define KPTS    27
#define GSTRIDE (KPTS * C_IO)   /* 1728 floats per G row */
#define BN_EPS  1e-5f

// ---------------------------------------------------------------------------
// Phase 0: zero the accumulator + BN-stats region of workspace
// ---------------------------------------------------------------------------
__global__ void zero_f32(float* __restrict__ p, long long n) {
  long long i = (long long)blockIdx.x * blockDim.x + threadIdx.x;
  long long stride = (long long)gridDim.x * blockDim.x;
  for (; i < n; i += stride) p[i] = 0.0f;
}

// ---------------------------------------------------------------------------
// Phase 1: G[r, kp*64+o] = sum_c feat[r,c] * W[kp,c,o]
// grid = (ceil(nref/128), 27), block = 256 (8 waves). Each block stages
// W[kp] (64x64 f32, 16 KB) in LDS *in B-fragment order*: K-row pairs are
// interleaved, BsP[(kk>>1)*128 + c*2 + (kk&1)] = W[kk][c], so every WMMA B
// fragment is one aligned ds_load_b64 straight into an even VGPR pair
// (no repack movs). Each wave computes a 16x64 slab with 4 chained
// V_WMMA_F32_16X16X4_F32 accumulators over 16 K-steps.
// ---------------------------------------------------------------------------
__global__ __launch_bounds__(256)
void gemm_feat_w(const float* __restrict__ feat, const float* __restrict__ W,
                 float* __restrict__ G, int nref) {
  __shared__ float Bs[C_IO * C_IO];   // permuted W[kp]

  const int kp = blockIdx.y;
  const int t  = threadIdx.x;

  // cooperative fill: coalesced float4 global reads, interleaved LDS scatter
  const float4* wsrc = (const float4*)(W + (size_t)kp * C_IO * C_IO);
#pragma unroll
  for (int i = 0; i < 4; ++i) {
    int v = t + i * 256;            // float4 index, 0..1023
    float4 d = wsrc[v];
    int kk = v >> 4;                // source row (64 cols = 16 float4)
    int c0 = (v & 15) * 4;          // first source col
    float* dst = Bs + (kk >> 1) * 128 + (kk & 1) + c0 * 2;
    dst[0] = d.x; dst[2] = d.y; dst[4] = d.z; dst[6] = d.w;
  }
  __syncthreads();

  const int wave = t >> 5;
  const int lane = t & 31;
  const int hh   = lane >> 4;   // lane half selects K pair (A) / row pair (B)
  const int l15  = lane & 15;
  const int mrow = blockIdx.x * 128 + wave * 16;

  // A fragment source: feat[mrow + l15, k0 + hh*2 + {0,1}] (clamped row;
  // stores are guarded; WMMA needs full EXEC so no branching here)
  int arow = mrow + l15;
  if (arow >= nref) arow = nref - 1;
  const float* aptr = feat + (size_t)arow * C_IO + hh * 2;

  // B fragments: pair index p = k0/2 + hh, element pair at col nt*16+l15
  const float* bfrag = Bs + hh * 128 + l15 * 2;

  v8f acc0 = {}, acc1 = {}, acc2 = {}, acc3 = {};
#pragma unroll
  for (int k0 = 0; k0 < C_IO; k0 += 4) {
    v2f a = *(const v2f*)(aptr + k0);
    const float* bb = bfrag + (k0 >> 1) * 128;
    v2f b0 = *(const v2f*)(bb + 0);
    v2f b1 = *(const v2f*)(bb + 32);
    v2f b2 = *(const v2f*)(bb + 64);
    v2f b3 = *(const v2f*)(bb + 96);
    acc0 = __builtin_amdgcn_wmma_f32_16x16x4_f32(false, a, false, b0, (short)0, acc0, false, false);
    acc1 = __builtin_amdgcn_wmma_f32_16x16x4_f32(false, a, false, b1, (short)0, acc1, false, false);
    acc2 = __builtin_amdgcn_wmma_f32_16x16x4_f32(false, a, false, b2, (short)0, acc2, false, false);
    acc3 = __builtin_amdgcn_wmma_f32_16x16x4_f32(false, a, false, b3, (short)0, acc3, false, false);
  }

  // D layout: VGPR i holds rows (i, i+8) split by lane half; N = l15
#pragma unroll
  for (int i = 0; i < 8; ++i) {
    int r = mrow + i + hh * 8;
    if (r < nref) {
      float* gp = G + (size_t)r * GSTRIDE + kp * C_IO + l15;
      gp[0]  = acc0[i];
      gp[16] = acc1[i];
      gp[32] = acc2[i];
      gp[48] = acc3[i];
    }
  }
}

// ---------------------------------------------------------------------------
// Phase 2: per-edge scatter  acc[q,:] += w_e * G[r, k*64 : k*64+64]
// One wave per edge, lane owns 2 channels. acc is 25.6 MB -> L2-resident,
// so the f32 atomics resolve in L2. Prefetch next edge's G row.
// ---------------------------------------------------------------------------
__global__ __launch_bounds__(256)
void scatter_edges(const int* __restrict__ e_ref, const int* __restrict__ e_query,
                   const int* __restrict__ e_kernel, const float* __restrict__ e_weight,
                   const float* __restrict__ G, float* __restrict__ acc, int E) {
  const int gtid   = blockIdx.x * blockDim.x + threadIdx.x;
  const int lane   = gtid & 31;
  const int warp   = gtid >> 5;
  const int nwarps = (gridDim.x * blockDim.x) >> 5;

  for (int e = warp; e < E; e += nwarps) {
    int   r = e_ref[e];
    int   q = e_query[e];
    int   k = e_kernel[e];
    float w = e_weight[e];

    int en = e + nwarps;
    if (en < E) {  // prefetch next iteration's gather row into cache
      int rn = e_ref[en];
      int kn = e_kernel[en];
      __builtin_prefetch(G + (size_t)rn * GSTRIDE + kn * C_IO + lane * 2, 0, 0);
    }

    float2 g = *((const float2*)(G + (size_t)r * GSTRIDE + k * C_IO) + lane);
    float* op = acc + (size_t)q * C_IO + lane * 2;
    unsafeAtomicAdd(op + 0, w * g.x);
    unsafeAtomicAdd(op + 1, w * g.y);
  }
}

// ---------------------------------------------------------------------------
// Phase 3: per-channel sum / sum-of-squares over Q rows
// ---------------------------------------------------------------------------
__global__ __launch_bounds__(256)
void bn_stats(const float* __restrict__ acc, float* __restrict__ stats, int Q) {
  const int c      = threadIdx.x & (C_IO - 1);
  const int row0   = blockIdx.x * (blockDim.x / C_IO) + (threadIdx.x >> 6);
  const int stride = gridDim.x * (blockDim.x / C_IO);
  float s = 0.0f, s2 = 0.0f;
  for (int q = row0; q < Q; q += stride) {
    float v = acc[(size_t)q * C_IO + c];
    s  += v;
    s2 += v * v;
  }
  unsafeAtomicAdd(&stats[c], s);
  unsafeAtomicAdd(&stats[C_IO + c], s2);
}

// ---------------------------------------------------------------------------
// Phase 4: BatchNorm (batch stats) + ReLU -> d_out
// ---------------------------------------------------------------------------
__global__ __launch_bounds__(256)
void bn_relu(const float* __restrict__ acc, const float* __restrict__ stats,
             const float* __restrict__ gamma, const float* __restrict__ beta,
             float* __restrict__ out, int Q) {
  long long i = (long long)blockIdx.x * blockDim.x + threadIdx.x;
  long long n = (long long)Q * C_IO;
  if (i >= n) return;
  int   c    = (int)(i & (C_IO - 1));
  float invQ = 1.0f / (float)Q;
  float mean = stats[c] * invQ;
  float var  = stats[C_IO + c] * invQ - mean * mean;
  float v = (acc[i] - mean) * rsqrtf(var + BN_EPS) * gamma[c] + beta[c];
  out[i] = fmaxf(v, 0.0f);
}

// ---------------------------------------------------------------------------
extern "C" void kernel_launch(void* const* d_in, const int* in_sizes, int n_in,
                              void* d_out, int out_size, void* d_ws, size_t ws_size,
                              hipStream_t stream) {
  const float* feat   = (const float*)d_in[0];
  const int*   e_ref  = (const int*)  d_in[1];
  const int*   e_qry  = (const int*)  d_in[2];
  const int*   e_ker  = (const int*)  d_in[3];
  const float* e_w    = (const float*)d_in[4];
  const float* W      = (const float*)d_in[5];
  const float* gamma  = (const float*)d_in[6];
  const float* beta   = (const float*)d_in[7];

  const int nref = in_sizes[0] / C_IO;   // 100000
  const int E    = in_sizes[1];          // 2,000,000
  const int Q    = out_size / C_IO;      // 100000

  // workspace layout: [ G : nref*1728 f32 | acc : Q*64 f32 | stats : 128 f32 ]
  float* G     = (float*)d_ws;
  float* acc   = G + (size_t)nref * GSTRIDE;
  float* stats = acc + (size_t)Q * C_IO;

  zero_f32<<<8192, 256, 0, stream>>>(acc, (long long)Q * C_IO + 2 * C_IO);

  dim3 g1((nref + 127) / 128, KPTS);
  gemm_feat_w<<<g1, 256, 0, stream>>>(feat, W, G, nref);

  scatter_edges<<<8192, 256, 0, stream>>>(e_ref, e_qry, e_ker, e_w, G, acc, E);

  bn_stats<<<256, 256, 0, stream>>>(acc, stats, Q);

  bn_relu<<<(out_size + 255) / 256, 256, 0, stream>>>(acc, stats, gamma, beta,
                                                      (float*)d_out, Q);
}